// BodyModel_1726576854864
// MI455X (gfx1250) — compile-verified
//
#include <hip/hip_runtime.h>
#include <hip/hip_bf16.h>
#include <math.h>

typedef __attribute__((ext_vector_type(2))) float v2f;
typedef __attribute__((ext_vector_type(8))) float v8f;

#define NBATCH 512
#define NV     6890
#define NJ     24
#define NBETA  10
#define PF     207           // (NJ-1)*9
#define PFP    208           // padded K for WMMA (multiple of 4); pad column is ZERO
#define N3     (NV * 3)      // 20670
#define NT_N   1292          // ceil(20670/16)
#define NT_V   431           // ceil(6890/16)

// ---------------------------------------------------------------------------
// K1: v_shaped = v_template + betas @ shapedirs^T   (memory-bound, 10 MACs/elt)
// ---------------------------------------------------------------------------
__global__ void k_shape(const float* __restrict__ betas,
                        const float* __restrict__ vtmpl,
                        const float* __restrict__ sdirs,
                        float* __restrict__ vsh) {
    int i = blockIdx.x * blockDim.x + threadIdx.x;
    const int total = NBATCH * N3;
    if (i >= total) return;
    int b = i / N3;
    int r = i - b * N3;                  // v*3 + k
    const float* bb = betas + b * NBETA;
    const float* ss = sdirs + (size_t)r * NBETA;
    float acc = vtmpl[r];
#pragma unroll
    for (int l = 0; l < NBETA; ++l) acc = fmaf(bb[l], ss[l], acc);
    vsh[i] = acc;
}

// ---------------------------------------------------------------------------
// K2: J[b,j,:] = sum_v J_regressor[j,v] * v_shaped[b,v,:]   one wave per (b,j)
// ---------------------------------------------------------------------------
__global__ void k_joints(const float* __restrict__ jreg,
                         const float* __restrict__ vsh,
                         float* __restrict__ J) {
    int wave = (blockIdx.x * blockDim.x + threadIdx.x) >> 5;
    int lane = threadIdx.x & 31;
    if (wave >= NBATCH * NJ) return;     // wave-uniform
    int b = wave / NJ, j = wave - b * NJ;
    const float* jrow = jreg + (size_t)j * NV;
    const float* vb   = vsh + (size_t)b * N3;
    float a0 = 0.f, a1 = 0.f, a2 = 0.f;
    for (int v = lane; v < NV; v += 32) {
        float w = jrow[v];
        a0 = fmaf(w, vb[v * 3 + 0], a0);
        a1 = fmaf(w, vb[v * 3 + 1], a1);
        a2 = fmaf(w, vb[v * 3 + 2], a2);
    }
#pragma unroll
    for (int off = 16; off; off >>= 1) {
        a0 += __shfl_down(a0, off, 32);
        a1 += __shfl_down(a1, off, 32);
        a2 += __shfl_down(a2, off, 32);
    }
    if (lane == 0) {
        float* o = J + (size_t)(b * NJ + j) * 3;
        o[0] = a0; o[1] = a1; o[2] = a2;
    }
}

// ---------------------------------------------------------------------------
// K3: Rodrigues per (b,j) + pose_feature (padded to 208 with zero column)
// ---------------------------------------------------------------------------
__global__ void k_rodrigues(const float* __restrict__ pose,
                            float* __restrict__ rm,
                            float* __restrict__ pf) {
    int i = blockIdx.x * blockDim.x + threadIdx.x;
    if (i >= NBATCH * NJ) return;
    int b = i / NJ, j = i - b * NJ;
    float rx = pose[b * NJ * 3 + j * 3 + 0] + 1e-8f;
    float ry = pose[b * NJ * 3 + j * 3 + 1] + 1e-8f;
    float rz = pose[b * NJ * 3 + j * 3 + 2] + 1e-8f;
    float ang = sqrtf(rx * rx + ry * ry + rz * rz);
    float inv = 1.0f / ang;
    float ux = rx * inv, uy = ry * inv, uz = rz * inv;
    float c = cosf(ang), s = sinf(ang), t = 1.0f - c;
    float R[9] = { c + t * ux * ux,      t * ux * uy - s * uz, t * ux * uz + s * uy,
                   t * uy * ux + s * uz, c + t * uy * uy,      t * uy * uz - s * ux,
                   t * uz * ux - s * uy, t * uz * uy + s * ux, c + t * uz * uz };
    float* ro = rm + (size_t)i * 9;
#pragma unroll
    for (int e = 0; e < 9; ++e) ro[e] = R[e];
    if (j > 0) {
        float* po = pf + b * PFP + (j - 1) * 9;
#pragma unroll
        for (int e = 0; e < 9; ++e)
            po[e] = R[e] - ((e == 0 || e == 4 || e == 8) ? 1.0f : 0.0f);
    } else {
        pf[b * PFP + PF] = 0.0f;         // ZERO pad column -> B row 207 is don't-care
    }
}

// ---------------------------------------------------------------------------
// K4: kinematic chain (serial over 24 joints), posed_joints + rel_transforms
//     one wave per batch; lanes 0..15 own one element of the 4x4
// ---------------------------------------------------------------------------
__global__ void k_chain(const float* __restrict__ rm,
                        const float* __restrict__ J,
                        const int* __restrict__ parents,
                        float* __restrict__ relT,
                        float* __restrict__ pjoints) {
    int b = blockIdx.x;
    int t = threadIdx.x;                 // 32 threads
    __shared__ float ch[NJ][16];
    __shared__ float tm[16];
    __shared__ float sJ[NJ * 3];
    __shared__ int   sp[NJ];
    for (int i = t; i < NJ * 3; i += 32) sJ[i] = J[(size_t)b * NJ * 3 + i];
    if (t < NJ) sp[t] = parents[t];
    __syncthreads();
    for (int j = 0; j < NJ; ++j) {
        int par = sp[j];
        if (t < 16) {
            int r = t >> 2, c = t & 3;
            float v;
            if (r < 3 && c < 3)      v = rm[(size_t)(b * NJ + j) * 9 + r * 3 + c];
            else if (r < 3)          v = sJ[j * 3 + r] - (j > 0 ? sJ[par * 3 + r] : 0.0f);
            else                     v = (c == 3) ? 1.0f : 0.0f;
            tm[t] = v;
        }
        __syncthreads();
        if (t < 16) {
            int r = t >> 2, c = t & 3;
            float v;
            if (j == 0) v = tm[t];
            else {
                v = 0.0f;
#pragma unroll
                for (int k = 0; k < 4; ++k) v = fmaf(ch[par][r * 4 + k], tm[k * 4 + c], v);
            }
            ch[j][t] = v;
        }
        __syncthreads();
    }
    // posed joints = transforms[:, :, :3, 3]
    for (int i = t; i < NJ * 3; i += 32) {
        int j = i / 3, k = i - j * 3;
        pjoints[(size_t)b * NJ * 3 + i] = ch[j][k * 4 + 3];
    }
    // rel_transforms: last column -= transforms @ [J, 0]
    for (int i = t; i < NJ * 16; i += 32) {
        int j = i >> 4, e = i & 15, m = e >> 2, n = e & 3;
        float v = ch[j][e];
        if (n == 3)
            v -= ch[j][m * 4 + 0] * sJ[j * 3 + 0] +
                 ch[j][m * 4 + 1] * sJ[j * 3 + 1] +
                 ch[j][m * 4 + 2] * sJ[j * 3 + 2];
        relT[(size_t)b * NJ * 16 + i] = v;
    }
}

// ---------------------------------------------------------------------------
// K5: v_posed += pose_feature(512x208) @ posedirs(207x20670)  [fp32 WMMA]
//     Main K-loop (rows 0..203): purely linear addressing, unconditional
//     loads. The single step that can touch the pad row (k0=204, row 207) is
//     peeled, with its B row index clamped (contribution is 0 since the A pad
//     column is zero). Stores are the only EXEC-guarded ops.
// ---------------------------------------------------------------------------
__global__ void k_posegemm(const float* __restrict__ pf,
                           const float* __restrict__ pd,
                           float* __restrict__ vw) {
    __shared__ float sA[16 * PFP];       // 13,312 bytes, shared by 4 waves
    int tid = threadIdx.x;
    int mt = blockIdx.y;
    for (int i = tid; i < 16 * PFP; i += 128) {
        int r = i / PFP, k = i - r * PFP;
        sA[i] = pf[(mt * 16 + r) * PFP + k];
    }
    __syncthreads();

    int w = tid >> 5, lane = tid & 31;
    int nt = blockIdx.x * 4 + w;         // always < 1292 (323*4 exact)
    int half = lane >> 4, l15 = lane & 15;
    int col  = nt * 16 + l15;
    bool cok = col < N3;                 // true except 4 lanes (2 columns) of the last tile
    int colc = cok ? col : (N3 - 1);     // clamped load column (garbage never stored)
    const float* arow = sA + l15 * PFP;
    const float* pcol = pd + colc;       // column base; rows stride N3
    v8f acc = {};
#pragma unroll 4
    for (int k0 = 0; k0 < PF - 3; k0 += 4) {         // k0 = 0..200, rows <= 203
        int ka = k0 + 2 * half;                      // f32 fragment striping (ISA 7.12.2)
        v2f a, bb;
        a.x  = arow[ka];
        a.y  = arow[ka + 1];
        bb.x = pcol[(size_t)ka * N3];
        bb.y = pcol[(size_t)(ka + 1) * N3];
        if ((k0 & 15) == 0 && k0 + 16 < PF)          // uniform; 1 prefetch / 4 steps
            __builtin_prefetch(&pcol[(size_t)(k0 + 16 + 2 * half) * N3], 0, 1);
        acc = __builtin_amdgcn_wmma_f32_16x16x4_f32(
            false, a, false, bb, (short)0, acc, false, false);
    }
    {   // peeled final step: k0 = 204, rows 204..207 (row 207 clamped, A-pad==0)
        int ka  = 204 + 2 * half;                    // 204 or 206
        int kb1 = (ka + 1 < PF) ? (ka + 1) : (PF - 1);
        v2f a, bb;
        a.x  = arow[ka];
        a.y  = arow[ka + 1];
        bb.x = pcol[(size_t)ka  * N3];
        bb.y = pcol[(size_t)kb1 * N3];
        acc = __builtin_amdgcn_wmma_f32_16x16x4_f32(
            false, a, false, bb, (short)0, acc, false, false);
    }
    if (cok) {
#pragma unroll
        for (int r = 0; r < 8; ++r) {    // C: VGPR r -> rows r / r+8
            int row = mt * 16 + r + 8 * half;
            size_t idx = (size_t)row * N3 + col;
            vw[idx] += acc[r];           // in-place: v_shaped -> v_posed
        }
    }
}

// ---------------------------------------------------------------------------
// K6: T(16x16) = lbs_w tile(16x24) @ rel_transforms_b(24x16) [fp32 WMMA],
//     then apply 3x4 transform to v_posed -> verts. Loads clamped, not guarded.
// ---------------------------------------------------------------------------
__global__ void k_lbs(const float* __restrict__ lbsw,
                      const float* __restrict__ relT,
                      const float* __restrict__ vp,
                      float* __restrict__ verts) {
    __shared__ float sR[NJ * 16];        // 384 floats
    __shared__ float sT[4][256];         // one 16x16 T tile per wave
    int tid = threadIdx.x;
    int b = blockIdx.y;
    for (int i = tid; i < NJ * 16; i += 128) sR[i] = relT[(size_t)b * NJ * 16 + i];
    __syncthreads();

    int w = tid >> 5, lane = tid & 31;
    int vt = blockIdx.x * 4 + w;         // may be 431 (out of range) -> clamped loads
    int half = lane >> 4, l15 = lane & 15;
    int vrow = vt * 16 + l15;
    int vr   = (vrow < NV) ? vrow : (NV - 1);   // clamped: garbage rows never stored
    const float* wrow = lbsw + (size_t)vr * NJ;
    v8f acc = {};
#pragma unroll
    for (int k0 = 0; k0 < NJ; k0 += 4) {
        int ka = k0 + 2 * half;
        v2f a, bb;
        a.x  = wrow[ka];
        a.y  = wrow[ka + 1];
        bb.x = sR[ka * 16 + l15];
        bb.y = sR[(ka + 1) * 16 + l15];
        acc = __builtin_amdgcn_wmma_f32_16x16x4_f32(
            false, a, false, bb, (short)0, acc, false, false);
    }
#pragma unroll
    for (int r = 0; r < 8; ++r)
        sT[w][(r + 8 * half) * 16 + l15] = acc[r];
    __syncthreads();

    if (lane < 16) {
        int v = vt * 16 + lane;
        if (v < NV) {
            const float* T = &sT[w][lane * 16];
            const float* p = vp + ((size_t)b * NV + v) * 3;
            float x = p[0], y = p[1], z = p[2];
            float* o = verts + ((size_t)b * NV + v) * 3;
            o[0] = fmaf(T[0], x, fmaf(T[1], y, fmaf(T[2],  z, T[3])));
            o[1] = fmaf(T[4], x, fmaf(T[5], y, fmaf(T[6],  z, T[7])));
            o[2] = fmaf(T[8], x, fmaf(T[9], y, fmaf(T[10], z, T[11])));
        }
    }
}

// ---------------------------------------------------------------------------
extern "C" void kernel_launch(void* const* d_in, const int* in_sizes, int n_in,
                              void* d_out, int out_size, void* d_ws, size_t ws_size,
                              hipStream_t stream) {
    const float* betas   = (const float*)d_in[0];
    const float* pose    = (const float*)d_in[1];
    const float* vtmpl   = (const float*)d_in[2];
    const float* sdirs   = (const float*)d_in[3];
    const float* pdirs   = (const float*)d_in[4];
    const float* jreg    = (const float*)d_in[5];
    const float* lbsw    = (const float*)d_in[6];
    const int*   parents = (const int*)d_in[7];

    float* verts   = (float*)d_out;                        // B*V*3
    float* pjoints = (float*)d_out + (size_t)NBATCH * N3;  // B*NJ*3

    float* ws  = (float*)d_ws;
    float* vsh = ws;                                   // B*V*3   (v_shaped -> v_posed in place)
    float* pf  = vsh + (size_t)NBATCH * N3;            // B*208   pose_feature (padded)
    float* rm  = pf  + (size_t)NBATCH * PFP;           // B*NJ*9  rot mats
    float* Jb  = rm  + (size_t)NBATCH * NJ * 9;        // B*NJ*3  joints
    float* rT  = Jb  + (size_t)NBATCH * NJ * 3;        // B*NJ*16 rel transforms

    {   // K1
        int total = NBATCH * N3;
        k_shape<<<(total + 255) / 256, 256, 0, stream>>>(betas, vtmpl, sdirs, vsh);
    }
    {   // K2: one wave per (b,j) => 12288 waves, 4 waves/block
        k_joints<<<(NBATCH * NJ) / 4, 128, 0, stream>>>(jreg, vsh, Jb);
    }
    {   // K3
        k_rodrigues<<<(NBATCH * NJ + 255) / 256, 256, 0, stream>>>(pose, rm, pf);
    }
    {   // K4
        k_chain<<<NBATCH, 32, 0, stream>>>(rm, Jb, parents, rT, pjoints);
    }
    {   // K5: flagship GEMM (512x208)@(207x20670) on fp32 WMMA
        dim3 g(NT_N / 4, 32);
        k_posegemm<<<g, 128, 0, stream>>>(pf, pdirs, vsh);
    }
    {   // K6: LBS WMMA + vertex transform
        dim3 g((NT_V + 3) / 4, NBATCH);
        k_lbs<<<g, 128, 0, stream>>>(lbsw, rT, vsh, verts);
    }
}